// QAPDecoder_31851477467306
// MI455X (gfx1250) — compile-verified
//
#include <hip/hip_runtime.h>
#include <hip/hip_bf16.h>

// Problem constants (match reference)
#define N1_ 1001
#define K_  16
#define NEG_INF_ (-1e9f)

typedef __attribute__((ext_vector_type(2))) float v2f;
typedef __attribute__((ext_vector_type(8))) float v8f;

// ---------------------------------------------------------------------------
// Kernel 1: batched query GEMM  query[B,2] = ctx[B,4] @ Wq[4,2] + bq
// One wave (32 lanes) handles a tile of 16 batch rows via V_WMMA_F32_16X16X4_F32.
// A layout (32-bit 16x4, ISA 7.12.2): lanes 0-15 hold M=lane, VGPR0=K0,VGPR1=K1;
//                                     lanes 16-31 hold M=lane-16, VGPR0=K2,VGPR1=K3.
// B mirrored K-interleave (4x16): VGPR0 = rows {K0 | K2}, VGPR1 = {K1 | K3},
// N striped across lanes 0-15 / 16-31. Only N=0,1 are live (E=2), rest zero.
// D layout: element (M,N): M<8 -> VGPR M, lane N ; M>=8 -> VGPR M-8, lane N+16.
// Out-of-range rows are clamped to Btot-1 BEFORE the WMMA, so their D rows
// duplicate the last valid row -> clamped stores are idempotent (no branches).
// ---------------------------------------------------------------------------
__global__ __launch_bounds__(32) void qap_query_wmma(
    const float* __restrict__ capacity,
    const float* __restrict__ used_capacity,
    const int*   __restrict__ current_node,
    const float* __restrict__ psi_prime,
    const int*   __restrict__ step_p,
    const int*   __restrict__ ncust_p,
    const float* __restrict__ Wq,    // [4][2]
    const float* __restrict__ bq,    // [2]
    float*       __restrict__ qbuf,  // [B][2] (workspace)
    int Btot)
{
    const int lane = threadIdx.x;            // 0..31
    const int m16  = lane & 15;
    const int hi   = (lane >> 4) & 1;        // 0: K={0,1}, 1: K={2,3}
    int b = blockIdx.x * 16 + m16;
    b = (b < Btot) ? b : (Btot - 1);         // clamp keeps EXEC uniform

    // --- per-batch context (branch-free; EXEC must be all-ones at the WMMA) ---
    const float cap = capacity[b];
    const float rem = cap - used_capacity[b];
    const int   cur = current_node[b];
    const float prog = (float)step_p[0] / (float)ncust_p[0];
    const float2 pv = *(const float2*)(psi_prime + ((size_t)b * N1_ + (size_t)cur) * 2);
    const float p0 = (cur == 0) ? 0.0f : pv.x;   // zero amplitude at depot
    const float p1 = (cur == 0) ? 0.0f : pv.y;

    // A fragment: ctx = [p0, p1, rem/cap, progress]
    v2f a;
    a.x = hi ? (rem / cap) : p0;   // K = 2*hi
    a.y = hi ? prog        : p1;   // K = 2*hi + 1
    // B fragment: Wq[K][N], live only for N<2
    const int ncl = (m16 < 2) ? m16 : 0;
    const float w0 = Wq[(hi ? 2 : 0) * 2 + ncl];
    const float w1 = Wq[(hi ? 3 : 1) * 2 + ncl];
    v2f bb;
    bb.x = (m16 < 2) ? w0 : 0.0f;
    bb.y = (m16 < 2) ? w1 : 0.0f;

    v8f c = {};
    c = __builtin_amdgcn_wmma_f32_16x16x4_f32(
        /*neg_a=*/false, a, /*neg_b=*/false, bb,
        /*c_mod=*/(short)0, c, /*reuse_a=*/false, /*reuse_b=*/false);

    // Extract query columns N=0,1 (lanes 0,1 carry M=0..7; lanes 16,17 carry M=8..15)
    if (m16 < 2) {
        const float be = bq[m16];
        const int mbase = blockIdx.x * 16 + (hi ? 8 : 0);
#pragma unroll
        for (int r = 0; r < 8; ++r) {
            int bw = mbase + r;
            bw = (bw < Btot) ? bw : (Btot - 1);   // idempotent duplicate write
            qbuf[(size_t)bw * 2 + m16] = c[r] + be;
        }
    }
}

// ---------------------------------------------------------------------------
// Kernel 2: one block per batch row. Stage psi row in LDS via the CDNA5 async
// global->LDS path (no VGPR round-trip, ASYNCcnt-tracked), then compute mask +
// hybrid logits with LDS gathers, fix depot mask, masked log-softmax.
// ---------------------------------------------------------------------------
__global__ __launch_bounds__(256) void qap_decode(
    const unsigned char* __restrict__ visited,     // bool [B][N1]
    const float* __restrict__ demands,             // [B][N1]
    const float* __restrict__ capacity,            // [B]
    const float* __restrict__ used_capacity,       // [B]
    const int*   __restrict__ current_node,        // [B]
    const float* __restrict__ psi_prime,           // [B][N1][2]
    const int*   __restrict__ knn_indices,         // [B][N1][16]
    const float* __restrict__ lam_p,               // [1]
    const float* __restrict__ qbuf,                // [B][2]
    float*       __restrict__ out_logprobs,        // [B][N1]
    float*       __restrict__ out_mask)            // [B][N1] as 0/1 floats
{
    __shared__ float2 psiL[N1_];      // staged psi row (gather target -> ds_load_b64)
    __shared__ float  logitL[N1_];    // raw (unmasked) logits
    __shared__ float  maskL[N1_];     // 0/1
    __shared__ float  redF[8];
    __shared__ int    redI[8];
    __shared__ float  bMax, bLse;

    const int b    = blockIdx.x;
    const int tid  = threadIdx.x;
    const int lane = tid & 31;
    const int wave = tid >> 5;

    const float cap = capacity[b];
    const float rem = cap - used_capacity[b];
    const int   cur = current_node[b];
    const float lam = lam_p[0];
    const float q0  = qbuf[(size_t)b * 2 + 0];
    const float q1  = qbuf[(size_t)b * 2 + 1];

    // --- stage psi row into LDS: GLOBAL_LOAD_ASYNC_TO_LDS_B64 per lane ---
    {
        // LDS byte offset of psiL (addrspace(3) ptrtoint == offset from LDS base)
        __attribute__((address_space(3))) float2* lp3 =
            (__attribute__((address_space(3))) float2*)&psiL[0];
        const unsigned int ldsBase = (unsigned int)(size_t)lp3;
        const unsigned long long gBase =
            (unsigned long long)(size_t)(psi_prime + (size_t)b * (N1_ * 2));
        for (int i = tid; i < N1_; i += 256) {
            const unsigned int       la = ldsBase + (unsigned int)i * 8u;
            const unsigned long long ga = gBase + (unsigned long long)i * 8ull;
            asm volatile("global_load_async_to_lds_b64 %0, %1, off"
                         :: "v"(la), "v"(ga) : "memory");
        }
        asm volatile("s_wait_asynccnt 0" ::: "memory");
    }
    __syncthreads();

    const unsigned char* vrow = visited + (size_t)b * N1_;
    const float* drow = demands + (size_t)b * N1_;
    const int*   krow = knn_indices + (size_t)b * N1_ * K_;

    // --- per-node mask + logits; track any feasible customer ---
    int anyfeas = 0;
    for (int n = tid; n < N1_; n += 256) {
        // prefetch next strip of the kNN index stream (dominant HBM consumer)
        if (n + 256 < N1_) __builtin_prefetch(krow + (size_t)(n + 256) * K_, 0, 0);

        const bool m = (vrow[n] != 0) || (drow[n] > rem);
        const float2 p = psiL[n];
        // kNN gather: 4x int4 (B128) index loads, 16x ds_load_b64 psi gathers
        const int4* kp = (const int4*)(krow + (size_t)n * K_);
        float S0 = 0.0f, S1 = 0.0f;
#pragma unroll
        for (int j = 0; j < 4; ++j) {
            const int4 id = kp[j];
            float2 a0 = psiL[id.x], a1 = psiL[id.y], a2 = psiL[id.z], a3 = psiL[id.w];
            S0 += a0.x + a1.x + a2.x + a3.x;
            S1 += a0.y + a1.y + a2.y + a3.y;
        }
        const float knn_sim = (p.x * S0 + p.y * S1) * (1.0f / (float)K_);
        const float compat  = q0 * p.x + q1 * p.y;
        logitL[n] = (1.0f - lam) * compat + lam * knn_sim;
        maskL[n]  = m ? 1.0f : 0.0f;
        if (n >= 1 && !m) anyfeas = 1;
    }
    // has_cust: wave32 OR-reduce, then cross-wave
#pragma unroll
    for (int off = 16; off; off >>= 1) anyfeas |= __shfl_xor(anyfeas, off, 32);
    if (lane == 0) redI[wave] = anyfeas;
    __syncthreads();
    if (tid == 0) {
        int hc = 0;
#pragma unroll
        for (int w = 0; w < 8; ++w) hc |= redI[w];
        // reference OVERRIDES mask[:,0] = at_depot & has_cust
        maskL[0] = ((cur == 0) && hc) ? 1.0f : 0.0f;
    }
    __syncthreads();

    // --- masked max ---
    float mx = NEG_INF_;
    for (int n = tid; n < N1_; n += 256) {
        const float x = (maskL[n] != 0.0f) ? NEG_INF_ : logitL[n];
        mx = fmaxf(mx, x);
    }
#pragma unroll
    for (int off = 16; off; off >>= 1) mx = fmaxf(mx, __shfl_xor(mx, off, 32));
    if (lane == 0) redF[wave] = mx;
    __syncthreads();
    if (tid == 0) {
        float m2 = redF[0];
#pragma unroll
        for (int w = 1; w < 8; ++w) m2 = fmaxf(m2, redF[w]);
        bMax = m2;
    }
    __syncthreads();
    const float M = bMax;

    // --- sum of exp ---
    float se = 0.0f;
    for (int n = tid; n < N1_; n += 256) {
        const float x = (maskL[n] != 0.0f) ? NEG_INF_ : logitL[n];
        se += expf(x - M);
    }
#pragma unroll
    for (int off = 16; off; off >>= 1) se += __shfl_xor(se, off, 32);
    if (lane == 0) redF[wave] = se;
    __syncthreads();
    if (tid == 0) {
        float s2 = 0.0f;
#pragma unroll
        for (int w = 0; w < 8; ++w) s2 += redF[w];
        bLse = logf(s2);
    }
    __syncthreads();
    const float L = bLse;

    // --- write outputs ---
    float* lp = out_logprobs + (size_t)b * N1_;
    float* mo = out_mask + (size_t)b * N1_;
    for (int n = tid; n < N1_; n += 256) {
        const float x = (maskL[n] != 0.0f) ? NEG_INF_ : logitL[n];
        lp[n] = x - M - L;
        mo[n] = maskL[n];
    }
}

extern "C" void kernel_launch(void* const* d_in, const int* in_sizes, int n_in,
                              void* d_out, int out_size, void* d_ws, size_t ws_size,
                              hipStream_t stream) {
    const unsigned char* visited  = (const unsigned char*)d_in[0];
    const float* demands          = (const float*)d_in[1];
    const float* capacity         = (const float*)d_in[2];
    const float* used_capacity    = (const float*)d_in[3];
    const int*   current_node     = (const int*)d_in[4];
    const float* psi_prime        = (const float*)d_in[5];
    const int*   knn_indices      = (const int*)d_in[6];
    const int*   step_p           = (const int*)d_in[7];
    const int*   ncust_p          = (const int*)d_in[8];
    const float* Wq               = (const float*)d_in[9];
    const float* bq               = (const float*)d_in[10];
    const float* lam_p            = (const float*)d_in[11];

    const int Btot = in_sizes[2];              // capacity is [B]
    float* qbuf = (float*)d_ws;                // [B][2] scratch

    float* out_logprobs = (float*)d_out;
    float* out_mask     = out_logprobs + (size_t)Btot * N1_;

    const int tiles = (Btot + 15) / 16;
    qap_query_wmma<<<tiles, 32, 0, stream>>>(
        capacity, used_capacity, current_node, psi_prime,
        step_p, ncust_p, Wq, bq, qbuf, Btot);

    qap_decode<<<Btot, 256, 0, stream>>>(
        visited, demands, capacity, used_capacity, current_node,
        psi_prime, knn_indices, lam_p, qbuf, out_logprobs, out_mask);
}